// InitLayer_52793738002950
// MI455X (gfx1250) — compile-verified
//
#include <hip/hip_runtime.h>

typedef __attribute__((ext_vector_type(2))) float v2f;
typedef __attribute__((ext_vector_type(8))) float v8f;

#define NG 65536   // grid points
#define NA 256     // atoms
#define NB 2       // batch

// One wave handles a 16-grid-point tile (columns) for one batch b, looping
// over 16 atom tiles (rows). d2 via V_WMMA_F32_16X16X4_F32 using
//   (log2e*d)^2 = q_g + s_a - 2 (M c_g).r_a,   M = (log2e)^2 * W^T W
// A row (atom a):  (r0, r1, r2, s_a)
// B col (grid g):  (-2v0, -2v1, -2v2, 1),  v = M c_g
// C[.,n] = q_g[n] = c^T M c  (per-lane uniform, loop-invariant -> hoisted)
__global__ __launch_bounds__(256) void exp_dist_wmma_kernel(
    const float* __restrict__ R,      // [B, A, 3]
    const float* __restrict__ coords, // [G, 3]
    const float* __restrict__ W,      // [3, 3] row-major
    float* __restrict__ out)          // [B, G]
{
    const int lane  = threadIdx.x & 31;
    const int wave  = threadIdx.x >> 5;
    const int wid   = blockIdx.x * 8 + wave;      // global wave id, 8192 total
    const int b     = wid >> 12;                  // 4096 g-tiles per batch
    const int gbase = (wid & 4095) << 4;          // 16 grid points per wave

    // ---- M = (log2e)^2 * W^T W  (uniform) ----
    const float LOG2E = 1.44269504088896340736f;
    const float L2    = LOG2E * LOG2E;
    const float w00 = W[0], w01 = W[1], w02 = W[2];
    const float w10 = W[3], w11 = W[4], w12 = W[5];
    const float w20 = W[6], w21 = W[7], w22 = W[8];
    const float M00 = L2 * (w00*w00 + w10*w10 + w20*w20);
    const float M01 = L2 * (w00*w01 + w10*w11 + w20*w21);
    const float M02 = L2 * (w00*w02 + w10*w12 + w20*w22);
    const float M11 = L2 * (w01*w01 + w11*w11 + w21*w21);
    const float M12 = L2 * (w01*w02 + w11*w12 + w21*w22);
    const float M22 = L2 * (w02*w02 + w12*w12 + w22*w22);

    const int n     = lane & 15;   // B-column (grid) / A-row (atom) index
    const int khalf = lane >> 4;   // which K pair this lane holds

    // ---- B matrix (4 x 16 grid points), loop-invariant ----
    const int g = gbase + n;
    const float c0 = coords[3*g + 0], c1 = coords[3*g + 1], c2 = coords[3*g + 2];
    const float v0 = M00*c0 + M01*c1 + M02*c2;
    const float v1 = M01*c0 + M11*c1 + M12*c2;
    const float v2 = M02*c0 + M12*c1 + M22*c2;
    const float q  = c0*v0 + c1*v1 + c2*v2;     // c^T M c (per-lane uniform)
    v2f Bmat;
    Bmat.x = khalf ? (-2.0f*v2) : (-2.0f*v0);   // K=2 : K=0
    Bmat.y = khalf ? 1.0f       : (-2.0f*v1);   // K=3 : K=1

    const float* __restrict__ Rb = R + b * (NA * 3);

    float acc0 = 0.f, acc1 = 0.f;   // two chains to break add latency

    #pragma unroll 4
    for (int t = 0; t < 16; ++t) {
        // ---- A matrix (16 atoms x 4): row = lane&15 ----
        const int a = (t << 4) + n;
        const float r0 = Rb[3*a + 0], r1 = Rb[3*a + 1], r2 = Rb[3*a + 2];
        const float u0 = M00*r0 + M01*r1 + M02*r2;
        const float u1 = M01*r0 + M11*r1 + M12*r2;
        const float u2 = M02*r0 + M12*r1 + M22*r2;
        const float s  = r0*u0 + r1*u1 + r2*u2;  // r^T M r
        v2f Amat;
        Amat.x = khalf ? r2 : r0;                // K=2 : K=0
        Amat.y = khalf ? s  : r1;                // K=3 : K=1

        // C init: every element of this lane's column = q_g (uniform,
        // loop-invariant -> compiler keeps one broadcast copy outside loop)
        v8f Cm;
        #pragma unroll
        for (int i = 0; i < 8; ++i) Cm[i] = q;

        // (log2e*d)^2 tile = A x B + C   (16x16 f32)
        Cm = __builtin_amdgcn_wmma_f32_16x16x4_f32(
                /*neg_a=*/false, Amat, /*neg_b=*/false, Bmat,
                /*c_mod=*/(short)0, Cm, /*reuse_a=*/false, /*reuse_b=*/false);

        // exp(d) = exp2(log2e*d).  |.| handles FP-cancellation d2 < 0 and
        // folds into a free source modifier on v_sqrt_f32.
        #pragma unroll
        for (int i = 0; i < 8; ++i) {
            float d = __builtin_amdgcn_sqrtf(__builtin_fabsf(Cm[i]));
            float e = __builtin_amdgcn_exp2f(d);
            if (i & 1) acc1 += e; else acc0 += e;
        }
    }

    // lane L (<16) holds atom rows 0-7, lane L+16 holds rows 8-15, same g.
    float tot = acc0 + acc1;
    tot += __shfl_xor(tot, 16, 32);

    if (lane < 16) {
        out[b * NG + gbase + lane] = tot;        // coalesced 64B per wave
    }
}

extern "C" void kernel_launch(void* const* d_in, const int* in_sizes, int n_in,
                              void* d_out, int out_size, void* d_ws, size_t ws_size,
                              hipStream_t stream) {
    const float* R      = (const float*)d_in[0];  // [2, 256, 3]
    const float* coords = (const float*)d_in[1];  // [65536, 3]
    const float* W      = (const float*)d_in[2];  // [3, 3]
    // d_in[3] = Z (int32, unused by the reference math)
    float* out = (float*)d_out;                   // [2, 65536]

    // 8192 waves total = 1024 blocks x 8 waves
    exp_dist_wmma_kernel<<<1024, 256, 0, stream>>>(R, coords, W, out);
}